// TopkSparseAutoEncoder2Child_7456063225988
// MI455X (gfx1250) — compile-verified
//
#include <hip/hip_runtime.h>
#include <hip/hip_bf16.h>
#include <stddef.h>

// ---------------------------------------------------------------------------
// TopK Sparse AutoEncoder with 2 children — MI455X (gfx1250, wave32, WMMA)
//
// Memory-bound (453 MB fp32 weights @ 23.3 TB/s ~ 19.4 us floor): keep fp32,
// use V_WMMA_F32_16X16X4_F32 for encoder GEMMs, and stream weight tiles with
// CDNA5 async global->LDS DMA (ASYNCcnt) double-buffered under the WMMAs.
// Decoder exploits activation sparsity (150+75+75 of 24576 per row).
// ---------------------------------------------------------------------------

#define SEQ    4096
#define DMODEL 768
#define FDIM   24576
#define TOPK   150
#define CHILDK 75
#define ROWS   16      // sequence rows per workgroup (one WMMA M-tile)
#define NT     128     // features per outer tile (8 waves x 16)
#define KCH    16      // k-chunk staged in LDS (double buffered)
#define LSTR   20      // LDS row stride: 80B (16B aligned), 16 distinct banks

typedef float v2f __attribute__((ext_vector_type(2)));
typedef float v8f __attribute__((ext_vector_type(8)));

// ---------------- small fixed-capacity min-heap (exact top-k) ----------------
__device__ __forceinline__ void heap_sift_down(float* hv, int* hi, int n) {
  int i = 0;
  for (;;) {
    int l = 2 * i + 1, r = 2 * i + 2, sm = i;
    if (l < n && hv[l] < hv[sm]) sm = l;
    if (r < n && hv[r] < hv[sm]) sm = r;
    if (sm == i) break;
    float tv = hv[i]; hv[i] = hv[sm]; hv[sm] = tv;
    int   ti = hi[i]; hi[i] = hi[sm]; hi[sm] = ti;
    i = sm;
  }
}

__device__ __forceinline__ void heap_push(float* hv, int* hi, int& n, int cap,
                                          float v, int j) {
  if (n < cap) {
    int i = n++;
    hv[i] = v; hi[i] = j;
    while (i > 0) {
      int p = (i - 1) >> 1;
      if (hv[p] <= hv[i]) break;
      float tv = hv[p]; hv[p] = hv[i]; hv[i] = tv;
      int   ti = hi[p]; hi[p] = hi[i]; hi[i] = ti;
      i = p;
    }
  } else if (v > hv[0]) {
    hv[0] = v; hi[0] = j;
    heap_sift_down(hv, hi, cap);
  }
}

// ---------------- CDNA5 async global->LDS helpers (ASYNCcnt tracked) --------
__device__ __forceinline__ unsigned lds_addr32(const void* p) {
  // flat addresses into LDS carry the LDS byte offset in the low 32 bits
  return (unsigned)(size_t)p;
}

__device__ __forceinline__ void async_ld_b128(unsigned lds, const void* g) {
  asm volatile("global_load_async_to_lds_b128 %0, %1, off"
               :: "v"(lds), "v"((unsigned long long)(size_t)g) : "memory");
}

__device__ __forceinline__ void async_ld_b32(unsigned lds, const void* g) {
  asm volatile("global_load_async_to_lds_b32 %0, %1, off"
               :: "v"(lds), "v"((unsigned long long)(size_t)g) : "memory");
}

// Issue one k-chunk: every thread issues exactly 3 async ops (uniform EXEC),
// so per-wave ASYNCcnt bookkeeping is deterministic: 3 per chunk.
__device__ __forceinline__ void issue_chunk(
    const float* __restrict__ x, const float* __restrict__ W, int s0, int j0,
    int kc, int tid, float (*xs)[LSTR], float (*wt)[LSTR]) {
  // weight tile: 128 rows x 16 k = 512 x b128, 2 per thread
  for (int q = 0; q < 2; ++q) {
    const int idx = tid + q * 256;
    const int row = idx >> 2;
    const int c4  = (idx & 3) << 2;
    async_ld_b128(lds_addr32(&wt[row][c4]),
                  W + (size_t)(j0 + row) * DMODEL + kc + c4);
  }
  // x tile: 16 rows x 16 k = 256 x b32, 1 per thread
  {
    const int r = tid >> 4, c = tid & 15;
    async_ld_b32(lds_addr32(&xs[r][c]),
                 x + (size_t)(s0 + r) * DMODEL + kc + c);
  }
}

// ---------------- shared WMMA GEMM tile: pre[16 x 128] = x @ W^T + b ---------
// A (16x4 fp32): lane m<16 holds K=kk,kk+1 ; lane m+16 holds K=kk+2,kk+3
// B (4x16 fp32): lanes 0-15 hold N, VGPR0/1 = K rows {0,1}; lanes 16-31 = {2,3}
// C (16x16 f32): VGPR v -> M=v (lanes 0-15), M=v+8 (lanes 16-31), N=lane&15
// xs/wt are double buffers: [2*ROWS][LSTR] and [2*NT][LSTR].
__device__ __forceinline__ void gemm_tile(
    const float* __restrict__ x, const float* __restrict__ W,
    const float* __restrict__ bias, int s0, int j0, int tid,
    float (*xs)[LSTR], float (*wt)[LSTR], float (*pt)[NT + 4]) {
  const int lane = tid & 31;
  const int wv   = tid >> 5;          // wave 0..7 owns features [j0+16*wv, +16)
  const int m    = lane & 15;         // doubles as M (A) and N (B)
  const int h2   = (lane >> 4) << 1;  // 0 or 2: K sub-offset for upper half-wave
  const int nb   = wv * 16;

  // warm L2 for the next feature tile's weights (speculative, far temporal)
  if (j0 + NT < FDIM)
    __builtin_prefetch(W + (size_t)(j0 + NT) * DMODEL + tid * 4, 0, 0);

  v8f acc = {};
  constexpr int NCH = DMODEL / KCH;  // 48 chunks
  issue_chunk(x, W, s0, j0, 0, tid, xs, wt);  // prologue: chunk 0 -> buf 0
  for (int i = 0; i < NCH; ++i) {
    const int cb = i & 1;
    if (i + 1 < NCH) {
      issue_chunk(x, W, s0, j0, (i + 1) * KCH, tid,
                  xs + (1 - cb) * ROWS, wt + (1 - cb) * NT);
      // chunk i complete (in-order), chunk i+1 still in flight under the WMMAs
      asm volatile("s_wait_asynccnt 0x3" ::: "memory");
    } else {
      asm volatile("s_wait_asynccnt 0x0" ::: "memory");
    }
    __syncthreads();  // all waves' chunk-i DMA landed
    float (*xc)[LSTR] = xs + cb * ROWS;
    float (*wc)[LSTR] = wt + cb * NT;
    for (int kk = 0; kk < KCH; kk += 4) {
      v2f a, b;
      a.x = xc[m][kk + h2];
      a.y = xc[m][kk + h2 + 1];
      b.x = wc[nb + m][kk + h2];
      b.y = wc[nb + m][kk + h2 + 1];
      acc = __builtin_amdgcn_wmma_f32_16x16x4_f32(
          false, a, false, b, (short)0, acc, false, false);
    }
    __syncthreads();  // buffer cb free before chunk i+2 overwrites it
  }
  const float bv = bias[j0 + nb + m];
  const int   rb = (lane >> 4) << 3;  // 0 or 8
  for (int v = 0; v < 8; ++v) pt[v + rb][nb + m] = acc[v] + bv;
  __syncthreads();
}

// ---------------- kernel 1: main encoder + exact streaming top-150 ----------
__global__ __launch_bounds__(256) void k_encode_topk(
    const float* __restrict__ x, const float* __restrict__ We,
    const float* __restrict__ be, float* __restrict__ topVal,
    int* __restrict__ topIdx, unsigned* __restrict__ gateBm) {
  __shared__ __align__(16) float xs[2 * ROWS][LSTR];
  __shared__ __align__(16) float wt[2 * NT][LSTR];
  __shared__ float pt[ROWS][NT + 4];
  __shared__ float hv[ROWS][TOPK];
  __shared__ int   hi[ROWS][TOPK];
  const int tid = threadIdx.x;
  const int s0  = blockIdx.x * ROWS;
  int hn = 0;  // heap fill (meaningful for tid < ROWS)

  for (int t = 0; t < FDIM / NT; ++t) {
    const int j0 = t * NT;
    gemm_tile(x, We, be, s0, j0, tid, xs, wt, pt);
    if (tid < ROWS) {
      for (int col = 0; col < NT; ++col)
        heap_push(hv[tid], hi[tid], hn, TOPK, pt[tid][col], j0 + col);
    }
  }
  // zero this block's gate-bitmap rows (ws is poisoned), then publish
  for (int i = tid; i < ROWS * (FDIM / 32); i += 256)
    gateBm[(size_t)s0 * (FDIM / 32) + i] = 0u;
  __syncthreads();
  if (tid < ROWS) {
    const int s = s0 + tid;
    for (int i = 0; i < TOPK; ++i) {
      const int j = hi[tid][i];
      topVal[(size_t)s * TOPK + i] = hv[tid][i];
      topIdx[(size_t)s * TOPK + i] = j;
      atomicOr(&gateBm[(size_t)s * (FDIM / 32) + (j >> 5)], 1u << (j & 31));
    }
  }
}

// ---------------- kernel 2: child encoder, keep only gated values ----------
__global__ __launch_bounds__(256) void k_encode_child(
    const float* __restrict__ x, const float* __restrict__ Wec,
    const float* __restrict__ bec, const int* __restrict__ topIdx,
    const unsigned* __restrict__ gateBm, float* __restrict__ cval) {
  __shared__ __align__(16) float xs[2 * ROWS][LSTR];
  __shared__ __align__(16) float wt[2 * NT][LSTR];
  __shared__ float pt[ROWS][NT + 4];
  __shared__ int   tIdx[ROWS][TOPK];
  const int tid = threadIdx.x;
  const int s0  = blockIdx.x * ROWS;

  for (int i = tid; i < ROWS * TOPK; i += 256)
    tIdx[i / TOPK][i % TOPK] = topIdx[(size_t)s0 * TOPK + i];
  __syncthreads();

  for (int t = 0; t < FDIM / NT; ++t) {
    const int j0 = t * NT;
    gemm_tile(x, Wec, bec, s0, j0, tid, xs, wt, pt);
    if (tid < ROWS) {
      const int    s   = s0 + tid;
      const size_t bmb = (size_t)s * (FDIM / 32) + (j0 >> 5);
      const unsigned w0 = gateBm[bmb + 0], w1 = gateBm[bmb + 1];
      const unsigned w2 = gateBm[bmb + 2], w3 = gateBm[bmb + 3];
      for (int col = 0; col < NT; ++col) {
        const unsigned w = (col < 32) ? w0 : (col < 64) ? w1
                                           : (col < 96) ? w2 : w3;
        if ((w >> (col & 31)) & 1u) {
          const int   j = j0 + col;
          const float v = pt[tid][col];
          for (int i = 0; i < TOPK; ++i)
            if (tIdx[tid][i] == j) { cval[(size_t)s * TOPK + i] = v; break; }
        }
      }
    }
  }
}

// ---------------- kernel 3: per-row top-75 of the 150 gated values ----------
// Reference topk runs over 24426 zeros + 150 gated values, so any gated value
// <= 0 ranks below the zero filler and contributes nothing: clamp to 0.
__global__ __launch_bounds__(256) void k_child_topk(
    const float* __restrict__ cval1, const float* __restrict__ cval2,
    const int* __restrict__ topIdx, float* __restrict__ cselVal,
    int* __restrict__ cselIdx) {
  const int s = blockIdx.x * blockDim.x + threadIdx.x;
  if (s >= SEQ) return;
  for (int c = 0; c < 2; ++c) {
    const float* cv = c ? cval2 : cval1;
    float hv[CHILDK]; int hi[CHILDK]; int n = 0;
    for (int i = 0; i < TOPK; ++i)
      heap_push(hv, hi, n, CHILDK, cv[(size_t)s * TOPK + i],
                topIdx[(size_t)s * TOPK + i]);
    float* ov = cselVal + ((size_t)c * SEQ + s) * CHILDK;
    int*   oi = cselIdx + ((size_t)c * SEQ + s) * CHILDK;
    for (int i = 0; i < CHILDK; ++i) {
      const float v = hv[i];
      ov[i] = (v > 0.f) ? v : 0.f;
      oi[i] = hi[i];
    }
  }
}

// ---------------- kernel 4: transpose decoders so columns are contiguous ----
__global__ __launch_bounds__(256) void k_transpose(
    const float* __restrict__ Wd, const float* __restrict__ Wd1,
    const float* __restrict__ Wd2, float* __restrict__ WdT) {
  __shared__ float tile[32][33];
  const int j0 = blockIdx.x * 32;
  const int d0 = blockIdx.y * 32;
  const int z  = blockIdx.z;
  const float* W  = (z == 0) ? Wd : (z == 1) ? Wd1 : Wd2;
  float*       WT = WdT + (size_t)z * FDIM * DMODEL;
  const int tx = threadIdx.x, ty = threadIdx.y;
  for (int i = 0; i < 4; ++i)
    tile[ty + 8 * i][tx] = W[(size_t)(d0 + ty + 8 * i) * FDIM + (j0 + tx)];
  __syncthreads();
  for (int i = 0; i < 4; ++i)
    WT[(size_t)(j0 + ty + 8 * i) * DMODEL + (d0 + tx)] = tile[tx][ty + 8 * i];
}

// ---------------- kernel 5: sparse decoder (rank-1 gathers, ~240x FLOP cut) -
__global__ __launch_bounds__(256) void k_decode(
    const float* __restrict__ topVal, const int* __restrict__ topIdx,
    const float* __restrict__ cselVal, const int* __restrict__ cselIdx,
    const float* __restrict__ WdT, const float* __restrict__ bd,
    const float* __restrict__ bd1, const float* __restrict__ bd2,
    float* __restrict__ out) {
  const int tid = threadIdx.x;
  const int s0  = blockIdx.x * ROWS;
  const int d0 = tid, d1 = tid + 256, d2 = tid + 512;  // 768 = 3*256
  const float bias0 = bd[d0] + bd1[d0] + bd2[d0];
  const float bias1 = bd[d1] + bd1[d1] + bd2[d1];
  const float bias2 = bd[d2] + bd1[d2] + bd2[d2];
  for (int r = 0; r < ROWS; ++r) {
    const int s = s0 + r;
    float a0 = bias0, a1 = bias1, a2 = bias2;
    for (int i = 0; i < TOPK; ++i) {
      const float  v  = topVal[(size_t)s * TOPK + i];
      const float* wr = WdT + (size_t)topIdx[(size_t)s * TOPK + i] * DMODEL;
      a0 += v * wr[d0]; a1 += v * wr[d1]; a2 += v * wr[d2];
    }
    for (int c = 0; c < 2; ++c) {
      const float* base = WdT + (size_t)(1 + c) * FDIM * DMODEL;
      const size_t off  = ((size_t)c * SEQ + s) * CHILDK;
      for (int i = 0; i < CHILDK; ++i) {
        const float v = cselVal[off + i];
        if (v != 0.f) {
          const float* wr = base + (size_t)cselIdx[off + i] * DMODEL;
          a0 += v * wr[d0]; a1 += v * wr[d1]; a2 += v * wr[d2];
        }
      }
    }
    out[(size_t)s * DMODEL + d0] = a0;
    out[(size_t)s * DMODEL + d1] = a1;
    out[(size_t)s * DMODEL + d2] = a2;
  }
}

// ---------------------------------------------------------------------------
extern "C" void kernel_launch(void* const* d_in, const int* in_sizes, int n_in,
                              void* d_out, int out_size, void* d_ws,
                              size_t ws_size, hipStream_t stream) {
  (void)in_sizes; (void)n_in; (void)out_size; (void)ws_size;
  const float* x   = (const float*)d_in[0];
  const float* We  = (const float*)d_in[1];
  const float* be  = (const float*)d_in[2];
  const float* Wd  = (const float*)d_in[3];
  const float* bd  = (const float*)d_in[4];
  const float* We1 = (const float*)d_in[5];
  const float* be1 = (const float*)d_in[6];
  const float* Wd1 = (const float*)d_in[7];
  const float* bd1 = (const float*)d_in[8];
  const float* We2 = (const float*)d_in[9];
  const float* be2 = (const float*)d_in[10];
  const float* Wd2 = (const float*)d_in[11];
  const float* bd2 = (const float*)d_in[12];

  // workspace layout (~254 MB total)
  char* ws = (char*)d_ws;
  float*    topVal  = (float*)ws;     ws += (size_t)SEQ * TOPK * 4;
  int*      topIdx  = (int*)ws;       ws += (size_t)SEQ * TOPK * 4;
  unsigned* gateBm  = (unsigned*)ws;  ws += (size_t)SEQ * (FDIM / 32) * 4;
  float*    cval1   = (float*)ws;     ws += (size_t)SEQ * TOPK * 4;
  float*    cval2   = (float*)ws;     ws += (size_t)SEQ * TOPK * 4;
  float*    cselVal = (float*)ws;     ws += (size_t)2 * SEQ * CHILDK * 4;
  int*      cselIdx = (int*)ws;       ws += (size_t)2 * SEQ * CHILDK * 4;
  float*    WdT     = (float*)ws;     // 3 * FDIM * DMODEL floats

  k_encode_topk<<<SEQ / ROWS, 256, 0, stream>>>(x, We, be, topVal, topIdx,
                                                gateBm);
  k_encode_child<<<SEQ / ROWS, 256, 0, stream>>>(x, We1, be1, topIdx, gateBm,
                                                 cval1);
  k_encode_child<<<SEQ / ROWS, 256, 0, stream>>>(x, We2, be2, topIdx, gateBm,
                                                 cval2);
  k_child_topk<<<SEQ / 256, 256, 0, stream>>>(cval1, cval2, topIdx, cselVal,
                                              cselIdx);
  {
    dim3 tg(FDIM / 32, DMODEL / 32, 3), tb(32, 8, 1);
    k_transpose<<<tg, tb, 0, stream>>>(Wd, Wd1, Wd2, WdT);
  }
  k_decode<<<SEQ / ROWS, 256, 0, stream>>>(topVal, topIdx, cselVal, cselIdx,
                                           WdT, bd, bd1, bd2, (float*)d_out);
}